// GraphAttentionLayer_68547678044317
// MI455X (gfx1250) — compile-verified
//
#include <hip/hip_runtime.h>
#include <hip/hip_bf16.h>
#include <math.h>
#include <stdint.h>

typedef __attribute__((ext_vector_type(2))) float v2f;
typedef __attribute__((ext_vector_type(8))) float v8f;

#define B_ 16
#define N_ 1024
#define C_ 128
#define L_ 4
#define F_ 64
#define JC_ 128                 // j-chunk of h staged in LDS (128*64*4B = 32 KB)
#define ALPHA_ 0.2f
#define NEG_BIG_ -9.0e15f

// -------------------------------------------------------------------------
// Kernel 1: h[b,l,n,f] = sum_c V[b,n,c] * W[l,c,f]
// via V_WMMA_F32_16X16X4_F32. One wave computes a [16 x 64] strip of h.
// A-frag layout: lane(0..15) holds M=lane, K = {0,1}; lane(16..31): K = {2,3}.
// C/D layout: VGPR r, lanes 0-15 -> M=r, lanes 16-31 -> M=8+r, N=lane%16.
// -------------------------------------------------------------------------
__global__ void gat_proj(const float* __restrict__ V, const float* __restrict__ W,
                         float* __restrict__ h) {
  const int lane = threadIdx.x & 31;
  const int wave = threadIdx.x >> 5;
  const int half = lane >> 4;
  const int row  = lane & 15;
  const int l  = blockIdx.y;
  const int b  = blockIdx.z;
  const int i0 = (blockIdx.x * 4 + wave) * 16;

  const float* Vb = V + ((size_t)b * N_ + i0) * C_;
  const float* Wl = W + (size_t)l * C_ * F_;

  v8f acc[4] = {};
  for (int kk = 0; kk < C_; kk += 4) {
    const int ka = kk + 2 * half;
    v2f a;
    a.x = Vb[(size_t)row * C_ + ka];
    a.y = Vb[(size_t)row * C_ + ka + 1];
#pragma unroll
    for (int t = 0; t < 4; ++t) {
      v2f bf;
      bf.x = Wl[(size_t)ka * F_ + t * 16 + row];
      bf.y = Wl[(size_t)(ka + 1) * F_ + t * 16 + row];
      acc[t] = __builtin_amdgcn_wmma_f32_16x16x4_f32(
          false, a, false, bf, (short)0, acc[t], false, false);
    }
  }
  float* hout = h + (((size_t)b * L_ + l) * N_ + i0) * F_;
#pragma unroll
  for (int t = 0; t < 4; ++t)
#pragma unroll
    for (int r = 0; r < 8; ++r)
      hout[(size_t)(r + 8 * half) * F_ + t * 16 + row] = acc[t][r];
}

// -------------------------------------------------------------------------
// Kernel 2: f1[b,l,n] = h[b,l,n,:] . a1[l,:] ; same for f2.  One wave per row.
// -------------------------------------------------------------------------
__global__ void gat_f12(const float* __restrict__ h, const float* __restrict__ a1,
                        const float* __restrict__ a2, float* __restrict__ f1,
                        float* __restrict__ f2) {
  const int lane = threadIdx.x & 31;
  const int wave = threadIdx.x >> 5;
  const size_t rowid = (size_t)blockIdx.x * 8 + wave;   // over b*L*N rows
  const int l = (int)((rowid / N_) % L_);
  const float* hr  = h  + rowid * F_;
  const float* a1l = a1 + (size_t)l * F_;
  const float* a2l = a2 + (size_t)l * F_;
  float p1 = hr[lane] * a1l[lane] + hr[lane + 32] * a1l[lane + 32];
  float p2 = hr[lane] * a2l[lane] + hr[lane + 32] * a2l[lane + 32];
#pragma unroll
  for (int off = 16; off >= 1; off >>= 1) {
    p1 += __shfl_down(p1, off, 32);
    p2 += __shfl_down(p2, off, 32);
  }
  if (lane == 0) { f1[rowid] = p1; f2[rowid] = p2; }
}

// -------------------------------------------------------------------------
// Kernel 3: streaming (online) column softmax stats over axis i.
// Per column j: m[j] = max_i val(i,j); sinv[j] = 1 / sum_i exp(val - m).
// val(i,j) = A[b,i,l,j] > 0 ? lrelu(f1[i]+f2[j]) : NEG_BIG.
// Coalesced over j (innermost axis of A). f1 staged in LDS.
// -------------------------------------------------------------------------
__global__ void gat_colstats(const float* __restrict__ A, const float* __restrict__ f1,
                             const float* __restrict__ f2, float* __restrict__ mcol,
                             float* __restrict__ sinv) {
  __shared__ float sf1[N_];
  const int tid = threadIdx.x;
  const int l = blockIdx.y, b = blockIdx.z;
  const int j = blockIdx.x * 256 + tid;
  const float* f1p = f1 + ((size_t)b * L_ + l) * N_;
  for (int i = tid; i < N_; i += 256) sf1[i] = f1p[i];
  __syncthreads();

  const float f2j = f2[((size_t)b * L_ + l) * N_ + j];
  const float* Ap = A + ((size_t)b * N_ * L_ + l) * N_ + j;  // stride i = L_*N_
  float mmax = -INFINITY, ssum = 0.f;
  for (int i = 0; i < N_; ++i) {
    float mask = Ap[(size_t)i * (L_ * N_)];
    float x = sf1[i] + f2j;
    float v = (x > 0.f) ? x : ALPHA_ * x;
    float val = (mask > 0.f) ? v : NEG_BIG_;
    float mn = fmaxf(mmax, val);
    ssum = ssum * __expf(mmax - mn) + __expf(val - mn);
    mmax = mn;
  }
  const size_t idx = ((size_t)b * L_ + l) * N_ + j;
  mcol[idx] = mmax;
  sinv[idx] = 1.0f / ssum;   // fully-masked column -> 1/N, matching reference
}

// -------------------------------------------------------------------------
// Kernel 4: out[b,i,f] = bias[f] + sum_l sum_j p(i,j) * h[b,l,j,f].
// 8 waves per block share the same b: h chunks are async-staged into LDS
// (global_load_async_to_lds_b128 -> s_wait_asynccnt 0 -> barrier) and
// B-fragments are read from LDS. p is computed per lane on the fly and fed
// straight into the WMMA A-fragment (i = lane%16, j = j0 + 2*(lane/16)+{0,1}).
// -------------------------------------------------------------------------
__global__ __launch_bounds__(256) void gat_out(
    const float* __restrict__ A, const float* __restrict__ h,
    const float* __restrict__ f1, const float* __restrict__ f2,
    const float* __restrict__ mcol, const float* __restrict__ sinv,
    const float* __restrict__ bias, float* __restrict__ out) {
  __shared__ float shH[JC_ * F_];                  // 32 KB staged h chunk
  __shared__ float shF2[N_], shM[N_], shS[N_];     // 12 KB per-l softmax stats

  const int tid  = threadIdx.x;
  const int lane = tid & 31;
  const int wave = tid >> 5;
  const int half = lane >> 4;
  const int row  = lane & 15;
  const int b  = blockIdx.y;
  const int i0 = (blockIdx.x * 8 + wave) * 16;

  v8f acc[4] = {};
  for (int l = 0; l < L_; ++l) {
    const size_t blN = ((size_t)b * L_ + l) * N_;
    __syncthreads();                                // previous l's math done
    for (int j = tid; j < N_; j += 256) {
      shF2[j] = f2[blN + j];
      shM[j]  = mcol[blN + j];
      shS[j]  = sinv[blN + j];
    }
    const float f1i = f1[blN + i0 + row];
    const float* hp = h + blN * F_;
    const float* Ap = A + ((size_t)(b * N_ + i0 + row) * L_ + l) * N_;

    for (int jc = 0; jc < N_; jc += JC_) {
      __syncthreads();                              // shH consumers done
      // ---- async stage h[b,l, jc..jc+JC_, :] into LDS (no VGPR round-trip)
      {
        const float* gsrc = hp + (size_t)jc * F_;   // contiguous JC_*F_ floats
#pragma unroll
        for (int v = tid; v < (JC_ * F_) / 4; v += 256) {
          unsigned lds_off = (unsigned)(uintptr_t)(&shH[v * 4]);
          unsigned long long ga = (unsigned long long)(uintptr_t)(gsrc + v * 4);
          asm volatile("global_load_async_to_lds_b128 %0, %1, off"
                       :: "v"(lds_off), "v"(ga) : "memory");
        }
        asm volatile("s_wait_asynccnt 0x0" ::: "memory");
      }
      __syncthreads();                              // LDS writes visible to all

      for (int j0 = jc; j0 < jc + JC_; j0 += 4) {
        const int ja = j0 + 2 * half;               // this lane's two j's
        float mk0 = Ap[ja], mk1 = Ap[ja + 1];
        float x0 = f1i + shF2[ja];
        float x1 = f1i + shF2[ja + 1];
        float v0 = (x0 > 0.f) ? x0 : ALPHA_ * x0;
        float v1 = (x1 > 0.f) ? x1 : ALPHA_ * x1;
        float val0 = (mk0 > 0.f) ? v0 : NEG_BIG_;
        float val1 = (mk1 > 0.f) ? v1 : NEG_BIG_;
        v2f a;
        a.x = __expf(val0 - shM[ja])     * shS[ja];
        a.y = __expf(val1 - shM[ja + 1]) * shS[ja + 1];
        const int jl = ja - jc;
#pragma unroll
        for (int t = 0; t < 4; ++t) {
          v2f bf;
          bf.x = shH[jl * F_ + t * 16 + row];
          bf.y = shH[(jl + 1) * F_ + t * 16 + row];
          acc[t] = __builtin_amdgcn_wmma_f32_16x16x4_f32(
              false, a, false, bf, (short)0, acc[t], false, false);
        }
      }
    }
  }
  float* op = out + ((size_t)b * N_ + i0) * F_;
#pragma unroll
  for (int t = 0; t < 4; ++t)
#pragma unroll
    for (int r = 0; r < 8; ++r)
      op[(size_t)(r + 8 * half) * F_ + t * 16 + row] = acc[t][r] + bias[t * 16 + row];
}

// -------------------------------------------------------------------------
extern "C" void kernel_launch(void* const* d_in, const int* in_sizes, int n_in,
                              void* d_out, int out_size, void* d_ws, size_t ws_size,
                              hipStream_t stream) {
  const float* V    = (const float*)d_in[0];
  const float* A    = (const float*)d_in[1];
  const float* W    = (const float*)d_in[2];
  const float* a1   = (const float*)d_in[3];
  const float* a2   = (const float*)d_in[4];
  const float* bias = (const float*)d_in[5];
  float* out = (float*)d_out;

  // workspace layout (fp32): h[16 MB] | f1 | f2 | m | sinv  (~17.8 MB total)
  float* h  = (float*)d_ws;
  float* f1 = h  + (size_t)B_ * L_ * N_ * F_;
  float* f2 = f1 + (size_t)B_ * L_ * N_;
  float* mc = f2 + (size_t)B_ * L_ * N_;
  float* si = mc + (size_t)B_ * L_ * N_;

  gat_proj<<<dim3(N_ / 64, L_, B_), 128, 0, stream>>>(V, W, h);
  gat_f12<<<dim3((B_ * L_ * N_) / 8), 256, 0, stream>>>(h, a1, a2, f1, f2);
  gat_colstats<<<dim3(N_ / 256, L_, B_), 256, 0, stream>>>(A, f1, f2, mc, si);
  gat_out<<<dim3(N_ / 128, B_), 256, 0, stream>>>(A, h, f1, f2, mc, si, bias, out);
}